// MMAG_46789373723082
// MI455X (gfx1250) — compile-verified
//
#include <hip/hip_runtime.h>
#include <math.h>

typedef float v2f __attribute__((ext_vector_type(2)));
typedef float v8f __attribute__((ext_vector_type(8)));

#define LN_EPS 1e-5f

// ---------------------------------------------------------------------------
// Elementwise / utility kernels
// ---------------------------------------------------------------------------
__global__ void fill_f32_kernel(float* __restrict__ p, float v, int n) {
  int i = blockIdx.x * blockDim.x + threadIdx.x;
  if (i < n) p[i] = v;
}

__global__ void fill_i32_kernel(int* __restrict__ p, int v, int n) {
  int i = blockIdx.x * blockDim.x + threadIdx.x;
  if (i < n) p[i] = v;
}

// float atomic max via sign-split trick (order independent -> deterministic)
__device__ __forceinline__ void atomic_max_f32(float* addr, float val) {
  if (val >= 0.0f) atomicMax((int*)addr, __float_as_int(val));
  else             atomicMin((unsigned int*)addr, (unsigned int)__float_as_int(val));
}

// Per-edge scatter-max: agg[dst, :] = max(agg[dst, :], x[src, :])
__global__ void edge_max_kernel(const float* __restrict__ x, const int* __restrict__ ei,
                                float* __restrict__ agg, int E, int D) {
  int e = blockIdx.x;
  int d = blockIdx.y * blockDim.x + threadIdx.x;
  if (d >= D) return;
  int s = ei[e];
  int t = ei[E + e];
  atomic_max_f32(&agg[(size_t)t * D + d], x[(size_t)s * D + d]);
}

// PyG fills empty segments (still -inf) with 0
__global__ void fixup_neginf_kernel(float* __restrict__ p, int n) {
  int i = blockIdx.x * blockDim.x + threadIdx.x;
  if (i < n) {
    float v = p[i];
    if (!__builtin_isfinite(v)) p[i] = 0.0f;
  }
}

// ---------------------------------------------------------------------------
// fp32 WMMA GEMM:  C[M,N] = (acc ? C : 0) + A @ W^T (+ bias)
//   A accessed as A[row, k] with optional idx0/idx1 row gather and a K-split
//   (fuses z = [x[src], x[dst]] without materializing z).  W is [N, K].
// Block tile 64x128, BK=16, double-buffered LDS with register prefetch.
// 8 waves, each computing a 32x32 slab = 2x2 WMMA fragments
// (16 x V_WMMA_F32_16X16X4_F32 per K stage).
// ---------------------------------------------------------------------------
__global__ __launch_bounds__(256) void gemm_f32_wmma_kernel(
    float* __restrict__ Cm, const float* __restrict__ A,
    const float* __restrict__ W, const float* __restrict__ bias,
    const int* __restrict__ idx0, const int* __restrict__ idx1,
    int M, int N, int K, int lda, int k_split, int accumulate)
{
  // 20 floats/row pad: 20*m mod 64 strides by 4 -> conflict-free, 16B-aligned rows
  __shared__ __align__(16) float As[2][64][20];
  __shared__ __align__(16) float Bs[2][128][20];

  const int t    = threadIdx.x;
  const int lane = t & 31;
  const int wave = t >> 5;
  const int lo   = lane & 15;   // M/N index within fragment
  const int hi   = lane >> 4;   // K-pair / row-half select
  const int wm   = wave & 1;    // 2 x 32-row slabs
  const int wn   = wave >> 1;   // 4 x 32-col slabs

  const int block_m = blockIdx.y * 64;
  const int block_n = blockIdx.x * 128;

  // A loader: each thread one float4  (row t>>2, k (t&3)*4)
  const int a_lrow = t >> 2;
  const int a_lk   = (t & 3) << 2;
  const int  am    = block_m + a_lrow;
  const bool a_ok  = am < M;
  const int  am_c  = a_ok ? am : 0;

  const float* base0;
  const float* base1 = nullptr;
  if (idx0) {
    base0 = A + (size_t)idx0[am_c] * lda;
    if (idx1) base1 = A + (size_t)idx1[am_c] * lda - k_split;
  } else {
    base0 = A + (size_t)am_c * lda;
  }

  // B loader: each thread two float4 (row t>>1, k (t&1)*8)
  const int b_lrow = t >> 1;
  const int b_lk   = (t & 1) << 3;
  const int  wrow  = block_n + b_lrow;
  const bool w_ok  = wrow < N;
  const float* wbase = W + (size_t)(w_ok ? wrow : 0) * K;

  v8f c00 = {}, c01 = {}, c10 = {}, c11 = {};

  float4 av, wv0, wv1;
  auto load_tiles = [&](int kb) {
    const int kga = kb + a_lk;
    av = make_float4(0.f, 0.f, 0.f, 0.f);
    if (a_ok) {
      const float* ap = (base1 && kga >= k_split) ? (base1 + kga) : (base0 + kga);
      av = *(const float4*)ap;
    }
    const int kgb = kb + b_lk;
    wv0 = make_float4(0.f, 0.f, 0.f, 0.f);
    wv1 = wv0;
    if (w_ok) {
      wv0 = *(const float4*)(wbase + kgb);
      wv1 = *(const float4*)(wbase + kgb + 4);
    }
  };
  auto store_tiles = [&](int buf) {
    *(float4*)&As[buf][a_lrow][a_lk]     = av;
    *(float4*)&Bs[buf][b_lrow][b_lk]     = wv0;
    *(float4*)&Bs[buf][b_lrow][b_lk + 4] = wv1;
  };

  load_tiles(0);
  store_tiles(0);
  __syncthreads();

  int cur = 0;
  for (int kb = 0; kb < K; kb += 16) {
    const bool has_next = (kb + 16) < K;
    if (has_next) load_tiles(kb + 16);   // global loads overlap the WMMA burst

#pragma unroll
    for (int k0 = 0; k0 < 16; k0 += 4) {
      const int kk = k0 + 2 * hi;  // lanes 0-15: K=k0,k0+1; lanes 16-31: K=k0+2,k0+3
      v2f a0 = *(const v2f*)&As[cur][wm * 32 + lo][kk];
      v2f a1 = *(const v2f*)&As[cur][wm * 32 + 16 + lo][kk];
      v2f b0 = *(const v2f*)&Bs[cur][wn * 32 + lo][kk];
      v2f b1 = *(const v2f*)&Bs[cur][wn * 32 + 16 + lo][kk];
      c00 = __builtin_amdgcn_wmma_f32_16x16x4_f32(false, a0, false, b0, (short)0, c00, false, false);
      c01 = __builtin_amdgcn_wmma_f32_16x16x4_f32(false, a0, false, b1, (short)0, c01, false, false);
      c10 = __builtin_amdgcn_wmma_f32_16x16x4_f32(false, a1, false, b0, (short)0, c10, false, false);
      c11 = __builtin_amdgcn_wmma_f32_16x16x4_f32(false, a1, false, b1, (short)0, c11, false, false);
    }
    if (has_next) {
      // safe: previous barrier guaranteed all reads of buffer (1-cur) finished
      store_tiles(1 - cur);
      __syncthreads();
      cur ^= 1;
    }
  }

  // C layout: VGPR r -> row hi*8 + r, col lo
  const int n0 = block_n + wn * 32 + lo;
  const int n1 = n0 + 16;
  const float bias0 = (bias != nullptr && n0 < N) ? bias[n0] : 0.f;
  const float bias1 = (bias != nullptr && n1 < N) ? bias[n1] : 0.f;
#pragma unroll
  for (int r = 0; r < 8; ++r) {
    const int m0 = block_m + wm * 32 + hi * 8 + r;
    const int m1 = m0 + 16;
    if (m0 < M) {
      if (n0 < N) {
        const size_t o = (size_t)m0 * N + n0;
        float v = c00[r] + bias0;
        if (accumulate) v += Cm[o];
        Cm[o] = v;
      }
      if (n1 < N) {
        const size_t o = (size_t)m0 * N + n1;
        float v = c01[r] + bias1;
        if (accumulate) v += Cm[o];
        Cm[o] = v;
      }
    }
    if (m1 < M) {
      if (n0 < N) {
        const size_t o = (size_t)m1 * N + n0;
        float v = c10[r] + bias0;
        if (accumulate) v += Cm[o];
        Cm[o] = v;
      }
      if (n1 < N) {
        const size_t o = (size_t)m1 * N + n1;
        float v = c11[r] + bias1;
        if (accumulate) v += Cm[o];
        Cm[o] = v;
      }
    }
  }
}

// ---------------------------------------------------------------------------
// BatchNorm column statistics (mean over axis 0), coalesced 64 cols / block
// ---------------------------------------------------------------------------
__global__ __launch_bounds__(256) void colstats_kernel(
    const float* __restrict__ X, float* __restrict__ mean, float* __restrict__ rstd,
    int M, int N)
{
  __shared__ float ss[4][64];
  __shared__ float sq[4][64];
  const int c   = threadIdx.x & 63;
  const int rg  = threadIdx.x >> 6;
  const int col = blockIdx.x * 64 + c;
  float s = 0.f, q = 0.f;
  if (col < N) {
    for (int r = rg; r < M; r += 4) {
      float v = X[(size_t)r * N + col];
      s += v;
      q += v * v;
    }
  }
  ss[rg][c] = s;
  sq[rg][c] = q;
  __syncthreads();
  if (rg == 0 && col < N) {
    float st = ss[0][c] + ss[1][c] + ss[2][c] + ss[3][c];
    float qt = sq[0][c] + sq[1][c] + sq[2][c] + sq[3][c];
    float mu  = st / (float)M;
    float var = qt / (float)M - mu * mu;
    mean[col] = mu;
    rstd[col] = rsqrtf(var + LN_EPS);
  }
}

__global__ void bn_apply_kernel(const float* __restrict__ X, float* __restrict__ Y,
                                const float* __restrict__ mean, const float* __restrict__ rstd,
                                const float* __restrict__ g, const float* __restrict__ b,
                                int total, int N) {
  int i = blockIdx.x * blockDim.x + threadIdx.x;
  if (i < total) {
    int c = i % N;
    Y[i] = (X[i] - mean[c]) * rstd[c] * g[c] + b[c];
  }
}

// x[:, :D] = bn ; x[:, D:2D] = bn
__global__ void build_x_kernel(const float* __restrict__ bn, float* __restrict__ x,
                               int total, int D) {
  int i = blockIdx.x * blockDim.x + threadIdx.x;
  if (i < total) {
    int r = i / D, c = i % D;
    float v = bn[i];
    x[(size_t)r * 2 * D + c]     = v;
    x[(size_t)r * 2 * D + D + c] = v;
  }
}

// deterministic scatter: highest pair index p wins for each node
__global__ void winner_kernel(const int* __restrict__ idx, int* __restrict__ w, int P) {
  int p = blockIdx.x * blockDim.x + threadIdx.x;
  if (p < P) atomicMax(&w[idx[p]], p);
}

__global__ void apply_comp_kernel(float* __restrict__ x, const int* __restrict__ w,
                                  const float* __restrict__ comp, int total, int D) {
  int i = blockIdx.x * blockDim.x + threadIdx.x;
  if (i < total) {
    int node = i / D, c = i % D;
    int wp = w[node];
    if (wp >= 0) x[(size_t)node * 2 * D + c] = comp[(size_t)wp * D + c];
  }
}

// ---------------------------------------------------------------------------
// Per-row LayerNorm + ReLU (in place), one block per row
// ---------------------------------------------------------------------------
__global__ __launch_bounds__(256) void layernorm_relu_kernel(
    float* __restrict__ X, const float* __restrict__ g, const float* __restrict__ b, int N)
{
  __shared__ float red[256];
  float* xr = X + (size_t)blockIdx.x * N;
  const int t = threadIdx.x;

  float s = 0.f;
  for (int c = t; c < N; c += 256) s += xr[c];
  red[t] = s;
  __syncthreads();
  for (int st = 128; st > 0; st >>= 1) {
    if (t < st) red[t] += red[t + st];
    __syncthreads();
  }
  const float mu = red[0] / (float)N;
  __syncthreads();

  float vs = 0.f;
  for (int c = t; c < N; c += 256) {
    float d = xr[c] - mu;
    vs += d * d;
  }
  red[t] = vs;
  __syncthreads();
  for (int st = 128; st > 0; st >>= 1) {
    if (t < st) red[t] += red[t + st];
    __syncthreads();
  }
  const float rs = rsqrtf(red[0] / (float)N + LN_EPS);
  __syncthreads();

  for (int c = t; c < N; c += 256) {
    float y = (xr[c] - mu) * rs * g[c] + b[c];
    xr[c] = y > 0.f ? y : 0.f;
  }
}

// ---------------------------------------------------------------------------
// Host orchestration
// ---------------------------------------------------------------------------
extern "C" void kernel_launch(void* const* d_in, const int* in_sizes, int n_in,
                              void* d_out, int out_size, void* d_ws, size_t ws_size,
                              hipStream_t stream) {
  (void)n_in; (void)out_size; (void)ws_size;

  const float* x_feat = (const float*)d_in[0];
  const int*   edge_index = (const int*)d_in[1];
  const int*   pairs_idx  = (const int*)d_in[2];
  const float* h1c = (const float*)d_in[3];
  const float* h2c = (const float*)d_in[4];
  const float* W1l = (const float*)d_in[5];
  const float* b1l = (const float*)d_in[6];
  const float* W1r = (const float*)d_in[7];
  const float* bn1_g = (const float*)d_in[8];
  const float* bn1_b = (const float*)d_in[9];
  const float* W2l = (const float*)d_in[10];
  const float* b2l = (const float*)d_in[11];
  const float* W2r = (const float*)d_in[12];
  const float* bn_g = (const float*)d_in[13];
  const float* bn_b = (const float*)d_in[14];
  const float* Wp  = (const float*)d_in[15];
  const float* bp  = (const float*)d_in[16];
  const float* pln_g = (const float*)d_in[17];
  const float* pln_b = (const float*)d_in[18];
  const float* Wpc = (const float*)d_in[19];
  const float* bpc = (const float*)d_in[20];
  const float* We  = (const float*)d_in[21];
  const float* be  = (const float*)d_in[22];
  const float* Wnm = (const float*)d_in[23];
  const float* bnm = (const float*)d_in[24];
  const float* nln_g = (const float*)d_in[25];
  const float* nln_b = (const float*)d_in[26];
  const float* Wnc = (const float*)d_in[27];
  const float* bnc = (const float*)d_in[28];

  const int H   = in_sizes[6];           // 512
  const int IN  = in_sizes[5] / H;       // 256
  const int OUT = in_sizes[11];          // 512
  const int C   = in_sizes[20];          // 40
  const int CE  = in_sizes[22];          // 2
  const int Nn  = in_sizes[0] / IN;      // 10000
  const int E   = in_sizes[1] / 2;       // 160000
  const int P   = in_sizes[2] / 2;       // 20000
  const int D2  = 2 * OUT;               // 1024
  const int D4  = 4 * OUT;               // 2048

  // ---- output layout (return order, flat) ----
  float* out         = (float*)d_out;
  float* node_logit  = out;                              // [Nn, C]
  float* pairs_logit = node_logit + (size_t)Nn * C;      // [P, C]
  float* pairs_cons  = pairs_logit + (size_t)P * C;      // [P, CE]
  float* x_out       = pairs_cons + (size_t)P * CE;      // [Nn, 2*OUT]
  float* gnn_x       = x_out + (size_t)Nn * D2;          // [Nn, OUT]

  // ---- workspace layout ----
  char* wsb = (char*)d_ws;
  size_t off = 0;
  auto walloc = [&](size_t bytes) -> char* {
    char* r = wsb + off;
    off += (bytes + 255) & ~(size_t)255;
    return r;
  };
  float* agg   = (float*)walloc((size_t)Nn * H * 4);    // reused layer1 (IN) / layer2 (H)
  float* hbuf  = (float*)walloc((size_t)Nn * H * 4);
  float* bnb   = (float*)walloc((size_t)Nn * OUT * 4);
  float* x1b   = (float*)walloc((size_t)Nn * OUT * 4);
  float* zh    = (float*)walloc((size_t)P * D4 * 4);
  float* mean1 = (float*)walloc((size_t)H * 4);
  float* rstd1 = (float*)walloc((size_t)H * 4);
  float* mean2 = (float*)walloc((size_t)OUT * 4);
  float* rstd2 = (float*)walloc((size_t)OUT * 4);
  int*   wsrc  = (int*)walloc((size_t)Nn * 4);
  int*   wdst  = (int*)walloc((size_t)Nn * 4);

  auto gemm = [&](float* Cm, const float* A, const float* W, const float* bias,
                  int M, int Ncols, int K, int lda,
                  const int* i0, const int* i1, int ksplit, int acc) {
    dim3 grid((Ncols + 127) / 128, (M + 63) / 64);
    gemm_f32_wmma_kernel<<<grid, dim3(256), 0, stream>>>(
        Cm, A, W, bias, i0, i1, M, Ncols, K, lda, ksplit, acc);
  };
  auto ew = [](int n) { return dim3((n + 255) / 256); };

  // ================= SAGE layer 1 =================
  {
    const int n = Nn * IN;
    fill_f32_kernel<<<ew(n), 256, 0, stream>>>(agg, -INFINITY, n);
    edge_max_kernel<<<dim3(E, (IN + 255) / 256), 256, 0, stream>>>(x_feat, edge_index, agg, E, IN);
    fixup_neginf_kernel<<<ew(n), 256, 0, stream>>>(agg, n);
  }
  gemm(hbuf, agg,    W1l, b1l,    Nn, H, IN, IN, nullptr, nullptr, 0, 0);
  gemm(hbuf, x_feat, W1r, nullptr, Nn, H, IN, IN, nullptr, nullptr, 0, 1);
  colstats_kernel<<<dim3((H + 63) / 64), 256, 0, stream>>>(hbuf, mean1, rstd1, Nn, H);
  bn_apply_kernel<<<ew(Nn * H), 256, 0, stream>>>(hbuf, hbuf, mean1, rstd1, bn1_g, bn1_b, Nn * H, H);

  // ================= SAGE layer 2 =================
  {
    const int n = Nn * H;
    fill_f32_kernel<<<ew(n), 256, 0, stream>>>(agg, -INFINITY, n);
    edge_max_kernel<<<dim3(E, (H + 255) / 256), 256, 0, stream>>>(hbuf, edge_index, agg, E, H);
    fixup_neginf_kernel<<<ew(n), 256, 0, stream>>>(agg, n);
  }
  gemm(gnn_x, agg,  W2l, b2l,    Nn, OUT, H, H, nullptr, nullptr, 0, 0);
  gemm(gnn_x, hbuf, W2r, nullptr, Nn, OUT, H, H, nullptr, nullptr, 0, 1);
  colstats_kernel<<<dim3((OUT + 63) / 64), 256, 0, stream>>>(gnn_x, mean2, rstd2, Nn, OUT);
  bn_apply_kernel<<<ew(Nn * OUT), 256, 0, stream>>>(gnn_x, bnb, mean2, rstd2, bn_g, bn_b, Nn * OUT, OUT);

  // ================= assemble x (d_out) ===========
  build_x_kernel<<<ew(Nn * OUT), 256, 0, stream>>>(bnb, x_out, Nn * OUT, OUT);
  fill_i32_kernel<<<ew(Nn), 256, 0, stream>>>(wsrc, -1, Nn);
  winner_kernel<<<ew(P), 256, 0, stream>>>(pairs_idx, wsrc, P);
  apply_comp_kernel<<<ew(Nn * OUT), 256, 0, stream>>>(x_out, wsrc, h1c, Nn * OUT, OUT);
  fill_i32_kernel<<<ew(Nn), 256, 0, stream>>>(wdst, -1, Nn);
  winner_kernel<<<ew(P), 256, 0, stream>>>(pairs_idx + P, wdst, P);
  apply_comp_kernel<<<ew(Nn * OUT), 256, 0, stream>>>(x_out, wdst, h2c, Nn * OUT, OUT);

  // ================= pair head ====================
  // z = [x[src], x[dst]] gathered on the fly inside the GEMM A operand
  gemm(zh, x_out, Wp, bp, P, D4, D4, D2, pairs_idx, pairs_idx + P, D2, 0);
  layernorm_relu_kernel<<<dim3(P), 256, 0, stream>>>(zh, pln_g, pln_b, D4);
  gemm(pairs_logit, zh, Wpc, bpc, P, C,  D4, D4, nullptr, nullptr, 0, 0);
  gemm(pairs_cons,  zh, We,  be,  P, CE, D4, D4, nullptr, nullptr, 0, 0);

  // ================= node head ====================
  gemm(x1b, x_out, Wnm, bnm, Nn, OUT, D2, D2, nullptr, nullptr, 0, 0);
  layernorm_relu_kernel<<<dim3(Nn), 256, 0, stream>>>(x1b, nln_g, nln_b, OUT);
  gemm(node_logit, x1b, Wnc, bnc, Nn, C, OUT, OUT, nullptr, nullptr, 0, 0);
}